// GINConv_40346922779435
// MI455X (gfx1250) — compile-verified
//
#include <hip/hip_runtime.h>

typedef __attribute__((ext_vector_type(2))) float v2f;
typedef __attribute__((ext_vector_type(8))) float v8f;

#define FEAT_IN  64
#define FEAT_OUT 128

// ---------------------------------------------------------------------------
// Kernel 1: h = (1 + eps) * x  (eps = 0 -> straight copy), float4 vectorized.
// ---------------------------------------------------------------------------
__global__ void gin_init_h(const float* __restrict__ x, float* __restrict__ h, int n4) {
    int i = blockIdx.x * blockDim.x + threadIdx.x;
    if (i < n4) {
        ((float4*)h)[i] = ((const float4*)x)[i];
    }
}

// ---------------------------------------------------------------------------
// Kernel 2: h[dst[e]] += x[src[e]]  (GIN sum aggregation).
// One thread per (edge, channel). Within a wave32, all 32 lanes share one
// edge and cover 32 consecutive channels: src/dst loads coalesce to a single
// broadcast, the x-row read is one contiguous 128B transaction, and the
// GLOBAL_ATOMIC_ADD_F32s hit consecutive addresses (L2-resident).
// ---------------------------------------------------------------------------
__global__ void gin_scatter_add(const float* __restrict__ x,
                                const int* __restrict__ src,
                                const int* __restrict__ dst,
                                float* __restrict__ h,
                                long long total) {
    long long i = (long long)blockIdx.x * blockDim.x + threadIdx.x;
    if (i >= total) return;
    int e = (int)(i >> 6);       // edge index
    int c = (int)(i & 63);       // channel 0..63
    int s = src[e];
    int d = dst[e];
    float v = x[(long long)s * FEAT_IN + c];
    atomicAdd(&h[(long long)d * FEAT_IN + c], v);
}

// ---------------------------------------------------------------------------
// Kernel 3: out = relu(h @ W + b) via V_WMMA_F32_16X16X4_F32 (exact fp32).
// Block = 256 threads = 8 waves. Block owns a 16-row stripe of h; wave w owns
// output columns [16w, 16w+16). Chain 16 WMMAs over K=64.
//
// A (16x4 f32): lane = m + 16*(k>=2), vgpr = k&1  -> contiguous float2 load.
// B (4x16 f32): lane = n + 16*(k>=2), vgpr = k&1  -> two W loads stride 128.
// C/D (16x16):  vgpr r -> row = r + 8*(lane>=16), col = lane&15.
// ---------------------------------------------------------------------------
__global__ __launch_bounds__(256)
void gin_gemm_wmma(const float* __restrict__ h,
                   const float* __restrict__ W,
                   const float* __restrict__ bias,
                   float* __restrict__ out) {
    const int wave  = threadIdx.x >> 5;      // 0..7
    const int lane  = threadIdx.x & 31;
    const int m0    = blockIdx.x * 16;       // row stripe (exact: 100000/16)
    const int n0    = wave * 16;             // column tile
    const int lmod  = lane & 15;             // m for A, n for B/C
    const int khalf = (lane >> 4) * 2;       // 0 or 2

    const float* __restrict__ hrow = h + (long long)(m0 + lmod) * FEAT_IN;

    v8f c = {};
    #pragma unroll
    for (int k0 = 0; k0 < FEAT_IN; k0 += 4) {
        // A fragment: h[m0+lmod][k0+khalf .. k0+khalf+1]
        v2f a = *(const v2f*)(hrow + k0 + khalf);
        // B fragment: W[k0+khalf][n0+lmod], W[k0+khalf+1][n0+lmod]
        const float* wp = W + (long long)(k0 + khalf) * FEAT_OUT + n0 + lmod;
        v2f bf;
        bf.x = wp[0];
        bf.y = wp[FEAT_OUT];
        // 8 args: (neg_a, A, neg_b, B, c_mod, C, reuse_a, reuse_b)
        c = __builtin_amdgcn_wmma_f32_16x16x4_f32(false, a, false, bf,
                                                  (short)0, c, false, false);
    }

    const float bv = bias[n0 + lmod];
    const int rbase = m0 + ((lane >> 4) << 3);   // +8 rows for upper half-wave
    #pragma unroll
    for (int r = 0; r < 8; ++r) {
        float v = c[r] + bv;
        v = v > 0.0f ? v : 0.0f;                 // ReLU
        out[(long long)(rbase + r) * FEAT_OUT + n0 + lmod] = v;
    }
}

// ---------------------------------------------------------------------------
// Host-side launcher. Inputs: x, src, dst, W, b. Output: relu(h @ W + b).
// d_ws holds h (N_NODES * 64 floats = 25.6 MB).
// ---------------------------------------------------------------------------
extern "C" void kernel_launch(void* const* d_in, const int* in_sizes, int n_in,
                              void* d_out, int out_size, void* d_ws, size_t ws_size,
                              hipStream_t stream) {
    const float* x    = (const float*)d_in[0];
    const int*   src  = (const int*)d_in[1];
    const int*   dst  = (const int*)d_in[2];
    const float* W    = (const float*)d_in[3];
    const float* bias = (const float*)d_in[4];
    float*       out  = (float*)d_out;
    float*       h    = (float*)d_ws;

    const int n_nodes = in_sizes[0] / FEAT_IN;
    const int n_edges = in_sizes[1];

    // 1) h = x
    const int n4 = n_nodes * FEAT_IN / 4;
    gin_init_h<<<(n4 + 255) / 256, 256, 0, stream>>>(x, h, n4);

    // 2) h[dst] += x[src]
    const long long total = (long long)n_edges * FEAT_IN;
    const int sblocks = (int)((total + 255) / 256);
    gin_scatter_add<<<sblocks, 256, 0, stream>>>(x, src, dst, h, total);

    // 3) out = relu(h @ W + b)
    gin_gemm_wmma<<<n_nodes / 16, 256, 0, stream>>>(h, W, bias, out);
}